// DiT_87308095193386
// MI455X (gfx1250) — compile-verified
//
#include <hip/hip_runtime.h>
#include <hip/hip_bf16.h>
#include <math.h>

#define B_ 8
#define TOK_ 256
#define HID_ 128
#define LAT_ 512
#define OUTC_ 256
#define DEPTH_ 36
#define HEADS_ 8
#define HD_ 64
#define MLPH_ 2048
#define FREQ_ 256
#define GROUPS_ 8

typedef __attribute__((ext_vector_type(2))) float v2f;
typedef __attribute__((ext_vector_type(4))) float v4f;
typedef __attribute__((ext_vector_type(8))) float v8f;

// ---------------------------------------------------------------------------
// f32 WMMA: D(16x16,f32) = A(16x4,f32) * B(4x16,f32) + C
// ---------------------------------------------------------------------------
__device__ __forceinline__ v8f wmma_f32_16x16x4(v2f a, v2f b, v8f c) {
  return __builtin_amdgcn_wmma_f32_16x16x4_f32(false, a, false, b, (short)0, c,
                                               false, false);
}

// ---------------------------------------------------------------------------
// Generic batched WMMA GEMM: D = epi(A @ op(B) + bias)
//   WNK=true : B is (N,K) row-major, used as B^T  (weights)
//   WNK=false: B is (K,N) row-major               (P @ V case)
// One 16-row M tile per workgroup, staged into LDS via async global->LDS
// (ASYNCcnt path). Each wave owns up to TPW_ n-tiles; the n-tile loop is
// interleaved inside the k-loop and the B fragments are software-pipelined
// one k-step ahead in distinct registers so JC loads stay in flight while
// the current WMMAs execute.
// ---------------------------------------------------------------------------
#define KC_ 512
#define LDSLD_ (KC_ + 4) /* rows stay 16B-aligned; 516%64=4 spreads banks */
#define TPW_ 4
#define USE_ASYNC_STAGE 1

enum { E_NONE = 0, E_SIG = 1, E_GELU = 2, E_MUL = 3, E_GATED = 4, E_TOUT = 5 };

struct GemmArgs {
  const float* A;
  const float* Bm;
  const float* bias;   // may be null
  float* D;
  const float* mul;    // E_MUL: mask base
  const float* gate;   // E_GATED: per-(b,col) gate
  const float* res;    // E_GATED: residual base (read)
  int M, N, K;
  int lda, ldb, ldd;
  int zdiv;            // z -> (zo = z/zdiv, zi = z%zdiv)
  long aO, aI, bO, bI, dO, dI, mulO;
  float alpha;         // E_MUL scale
  int rowsPerB;        // rows per batch image (TOK)
  int gateStride;      // ada row stride
};

template <bool WNK>
__device__ __forceinline__ v2f loadB(const float* p, int k, long ldb) {
  if (WNK) {
    return *(const v2f*)(p + k);
  } else {
    v2f b;
    b.x = p[(long)k * ldb];
    b.y = p[(long)(k + 1) * ldb];
    return b;
  }
}

template <int JC, bool WNK>
__device__ __forceinline__ void kloop(const float* __restrict__ arow,
                                      const float* const* bp, long ldb, int kcur,
                                      v8f* acc) {
  v2f bnext[JC];
#pragma unroll
  for (int j = 0; j < JC; ++j) bnext[j] = loadB<WNK>(bp[j], 0, ldb);
#pragma unroll 2
  for (int k = 0; k < kcur; k += 4) {
    v2f a = *(const v2f*)(arow + k);
    v2f bcur[JC];
#pragma unroll
    for (int j = 0; j < JC; ++j) bcur[j] = bnext[j];
    // prefetch next k-step (clamped re-read when at the tail: branch-free)
    const int kn = (k + 4 < kcur) ? (k + 4) : 0;
#pragma unroll
    for (int j = 0; j < JC; ++j) bnext[j] = loadB<WNK>(bp[j], kn, ldb);
#pragma unroll
    for (int j = 0; j < JC; ++j) acc[j] = wmma_f32_16x16x4(a, bcur[j], acc[j]);
  }
}

template <bool WNK, int EPI>
__global__ __launch_bounds__(256) void gemm_kernel(GemmArgs g) {
  __shared__ float sA[16 * LDSLD_];
  const int z = blockIdx.z;
  const int zo = z / g.zdiv;
  const int zi = z - zo * g.zdiv;
  const float* A = g.A + (long)zo * g.aO + (long)zi * g.aI;
  const float* Bm = g.Bm + (long)zo * g.bO + (long)zi * g.bI;
  float* D = g.D + (long)zo * g.dO + (long)zi * g.dI;

  const int m0 = blockIdx.x << 4;
  const int lane = threadIdx.x & 31;
  // force scalar: guarantees s_cmp/s_cbranch (EXEC stays all-ones around WMMA)
  const int wave = __builtin_amdgcn_readfirstlane(threadIdx.x >> 5);
  const int nwaves = blockDim.x >> 5;
  const int l15 = lane & 15;
  const int khalf = (lane >> 4) << 1;  // lanes 16-31 handle K+2,K+3
  const int ntiles = g.N >> 4;
  const int tbase = blockIdx.y * nwaves * TPW_;

  // number of n-tiles this wave owns (wave-uniform scalar)
  const int rem = ntiles - tbase - wave;
  int jc = (rem <= 0) ? 0 : (rem + nwaves - 1) / nwaves;
  if (jc > TPW_) jc = TPW_;

  int n0s[TPW_];
#pragma unroll
  for (int j = 0; j < TPW_; ++j) n0s[j] = (tbase + j * nwaves + wave) << 4;

  const v8f zero = {0.f, 0.f, 0.f, 0.f, 0.f, 0.f, 0.f, 0.f};
  v8f acc[TPW_];
#pragma unroll
  for (int j = 0; j < TPW_; ++j) acc[j] = zero;

  for (int kc = 0; kc < g.K; kc += KC_) {
    const int kcur = (g.K - kc < KC_) ? (g.K - kc) : KC_;
    // --- stage A tile (16 x kcur) into LDS: async global->LDS, b128/quad ---
    {
      const int qperrow = kcur >> 2;
      const int quads = qperrow << 4;  // multiple of blockDim.x for all sizes
      for (int q = threadIdx.x; q < quads; q += blockDim.x) {
        const int rr = q / qperrow;
        const int cq = (q - rr * qperrow) << 2;
        const float* src = A + (long)(m0 + rr) * g.lda + kc + cq;
#if USE_ASYNC_STAGE
        unsigned ldsa = (unsigned)(uintptr_t)&sA[rr * LDSLD_ + cq];
        asm volatile("global_load_async_to_lds_b128 %0, %1, off"
                     :
                     : "v"(ldsa), "v"(src)
                     : "memory");
#else
        *(v4f*)&sA[rr * LDSLD_ + cq] = *(const v4f*)src;
#endif
      }
#if USE_ASYNC_STAGE
      asm volatile("s_wait_asynccnt 0x0" ::: "memory");
#endif
    }
    __syncthreads();

    const float* arow = sA + l15 * LDSLD_ + khalf;
    const float* bp[TPW_];
#pragma unroll
    for (int j = 0; j < TPW_; ++j) {
      if (WNK)
        bp[j] = Bm + (long)(n0s[j] + l15) * g.ldb + kc + khalf;
      else
        bp[j] = Bm + (long)(kc + khalf) * g.ldb + n0s[j] + l15;
    }

    if (jc == TPW_)
      kloop<TPW_, WNK>(arow, bp, g.ldb, kcur, acc);
    else if (jc == 2)
      kloop<2, WNK>(arow, bp, g.ldb, kcur, acc);
    else if (jc == 1)
      kloop<1, WNK>(arow, bp, g.ldb, kcur, acc);
    else if (jc == 3)
      kloop<3, WNK>(arow, bp, g.ldb, kcur, acc);
    __syncthreads();
  }

  // Epilogue. C/D layout: VGPR i -> row (m0 + i + 8*(lane>=16)), col n0+(lane&15)
  const int rbase = m0 + ((lane >> 4) << 3);
  for (int j = 0; j < jc; ++j) {
    const int col = n0s[j] + l15;
    const float bv = g.bias ? g.bias[col] : 0.f;
#pragma unroll
    for (int i = 0; i < 8; ++i) {
      const int row = rbase + i;
      float v = acc[j][i] + bv;
      if (EPI == E_SIG) {
        v = 1.f / (1.f + expf(-v));
      } else if (EPI == E_GELU) {
        v = 0.5f * v * (1.f + erff(v * 0.70710678118654752f));
      } else if (EPI == E_MUL) {
        v = v * g.alpha * g.mul[(long)zo * g.mulO + (long)row * g.N + col];
      }
      if (EPI == E_GATED) {
        const int bidx = row / g.rowsPerB;
        const float gt = g.gate[(long)bidx * g.gateStride + col];
        D[(long)row * g.ldd + col] = g.res[(long)row * g.ldd + col] + gt * v;
      } else if (EPI == E_TOUT) {
        const int bidx = row / g.rowsPerB;
        const int n = row - bidx * g.rowsPerB;
        D[(long)bidx * g.N * g.rowsPerB + (long)col * g.rowsPerB + n] = v;
      } else {
        D[(long)row * g.ldd + col] = v;
      }
    }
  }
}

// ---------------------------------------------------------------------------
// Elementwise / normalization helper kernels
// ---------------------------------------------------------------------------
__global__ void pe_kernel(const float* __restrict__ coords, float* __restrict__ pe) {
  const int bn = blockIdx.x;
  const int b = bn / TOK_, n = bn % TOK_;
  const int j = threadIdx.x;  // 0..255 -> LAT/2
  float enc = coords[((size_t)b * 3 + 0) * TOK_ + n] * 10000.f +
              coords[((size_t)b * 3 + 1) * TOK_ + n] * 100.f +
              coords[((size_t)b * 3 + 2) * TOK_ + n];
  float dv = expf((float)(2 * j) * (-logf(10000.f) / (float)LAT_));
  float ang = enc * dv;
  float* p = pe + (size_t)bn * LAT_ + 2 * j;
  p[0] = sinf(ang);
  p[1] = cosf(ang);
}

__global__ void ln_x_kernel(const float* __restrict__ x, const float* __restrict__ w,
                            const float* __restrict__ bb, float* __restrict__ h0) {
  __shared__ float red[128];
  const int bn = blockIdx.x;
  const int b = bn / TOK_, n = bn % TOK_;
  const int c = threadIdx.x;  // 128
  float v = x[((size_t)b * HID_ + c) * TOK_ + n];
  red[c] = v;
  __syncthreads();
  for (int s = 64; s > 0; s >>= 1) {
    if (c < s) red[c] += red[c + s];
    __syncthreads();
  }
  float m = red[0] * (1.f / HID_);
  __syncthreads();
  float d = v - m;
  red[c] = d * d;
  __syncthreads();
  for (int s = 64; s > 0; s >>= 1) {
    if (c < s) red[c] += red[c + s];
    __syncthreads();
  }
  float var = red[0] * (1.f / HID_);
  h0[(size_t)bn * HID_ + c] = d * rsqrtf(var + 1e-5f) * w[c] + bb[c];
}

__device__ __forceinline__ float redsum256(float v, float* red) {
  const int t = threadIdx.x;
  red[t] = v;
  __syncthreads();
  for (int s = 128; s > 0; s >>= 1) {
    if (t < s) red[t] += red[t + s];
    __syncthreads();
  }
  float r = red[0];
  __syncthreads();
  return r;
}

__global__ void ln_proj2_addpe_kernel(const float* __restrict__ in,
                                      const float* __restrict__ w,
                                      const float* __restrict__ bb,
                                      const float* __restrict__ pe,
                                      float* __restrict__ out) {
  __shared__ float red[256];
  const int r = blockIdx.x, t = threadIdx.x;
  const float* p = in + (size_t)r * LAT_;
  float v0 = p[t], v1 = p[t + 256];
  float m = redsum256(v0 + v1, red) * (1.f / LAT_);
  float d0 = v0 - m, d1 = v1 - m;
  float var = redsum256(d0 * d0 + d1 * d1, red) * (1.f / LAT_);
  float ri = rsqrtf(var + 1e-5f);
  const float* pp = pe + (size_t)r * LAT_;
  float* o = out + (size_t)r * LAT_;
  o[t] = d0 * ri * w[t] + bb[t] + pp[t];
  o[t + 256] = d1 * ri * w[t + 256] + bb[t + 256] + pp[t + 256];
}

__global__ void modln_kernel(const float* __restrict__ in, const float* __restrict__ ada,
                             int adaStride, int shOff, int scOff,
                             float* __restrict__ out) {
  __shared__ float red[256];
  const int r = blockIdx.x, t = threadIdx.x, b = r / TOK_;
  const float* p = in + (size_t)r * LAT_;
  float v0 = p[t], v1 = p[t + 256];
  float m = redsum256(v0 + v1, red) * (1.f / LAT_);
  float d0 = v0 - m, d1 = v1 - m;
  float var = redsum256(d0 * d0 + d1 * d1, red) * (1.f / LAT_);
  float ri = rsqrtf(var + 1e-6f);
  const float* ab = ada + (size_t)b * adaStride;
  float* o = out + (size_t)r * LAT_;
  o[t] = d0 * ri * (1.f + ab[scOff + t]) + ab[shOff + t];
  o[t + 256] = d1 * ri * (1.f + ab[scOff + t + 256]) + ab[shOff + t + 256];
}

__global__ void gn_kernel(const float* __restrict__ in, const float* __restrict__ w,
                          const float* __restrict__ bb, float* __restrict__ out) {
  __shared__ float r1[256], r2[256];
  const int b = blockIdx.x / GROUPS_, g = blockIdx.x % GROUPS_;
  const int t = threadIdx.x;
  const int GC = LAT_ / GROUPS_;  // 64
  const float* base = in + (size_t)b * TOK_ * LAT_ + g * GC;
  float s = 0.f, s2 = 0.f;
  for (int i = t; i < TOK_ * GC; i += 256) {
    const int n = i >> 6, cl = i & 63;
    float v = base[(size_t)n * LAT_ + cl];
    s += v;
    s2 += v * v;
  }
  r1[t] = s;
  r2[t] = s2;
  __syncthreads();
  for (int st = 128; st > 0; st >>= 1) {
    if (t < st) {
      r1[t] += r1[t + st];
      r2[t] += r2[t + st];
    }
    __syncthreads();
  }
  const float inv = 1.f / (float)(TOK_ * GC);
  float m = r1[0] * inv;
  float var = r2[0] * inv - m * m;
  float ri = rsqrtf(var + 1e-5f);
  __syncthreads();
  float* ob = out + (size_t)b * TOK_ * LAT_ + g * GC;
  for (int i = t; i < TOK_ * GC; i += 256) {
    const int n = i >> 6, cl = i & 63;
    const int ch = g * GC + cl;
    float v = base[(size_t)n * LAT_ + cl];
    ob[(size_t)n * LAT_ + cl] = (v - m) * ri * w[ch] + bb[ch];
  }
}

__global__ void softmax_kernel(float* __restrict__ S) {
  __shared__ float red[256];
  float* p = S + (size_t)blockIdx.x * TOK_;
  const int t = threadIdx.x;
  float v = p[t];
  red[t] = v;
  __syncthreads();
  for (int s = 128; s > 0; s >>= 1) {
    if (t < s) red[t] = fmaxf(red[t], red[t + s]);
    __syncthreads();
  }
  float mx = red[0];
  __syncthreads();
  float e = expf(v - mx);
  red[t] = e;
  __syncthreads();
  for (int s = 128; s > 0; s >>= 1) {
    if (t < s) red[t] += red[t + s];
    __syncthreads();
  }
  p[t] = e / red[0];
}

__global__ void temb_kernel(const float* __restrict__ t, float* __restrict__ emb) {
  const int b = blockIdx.x, j = threadIdx.x;  // 128
  float f = expf(-logf(10000.f) * (float)j / 128.f);
  float a = t[b] * f;
  emb[(size_t)b * FREQ_ + j] = cosf(a);
  emb[(size_t)b * FREQ_ + 128 + j] = sinf(a);
}

// tiny M=8 GEMMs (t-embedder / adaLN): one thread per output element
__global__ void small_gemm_kernel(const float* __restrict__ X, const float* __restrict__ W,
                                  const float* __restrict__ bias, float* __restrict__ Y,
                                  int N, int K, int act) {
  const int n = blockIdx.x * blockDim.x + threadIdx.x;
  const int b = blockIdx.y;
  const float* x = X + (size_t)b * K;
  const float* w = W + (size_t)n * K;
  float acc = bias[n];
  for (int k = 0; k < K; ++k) acc += x[k] * w[k];
  if (act == 1) acc = acc / (1.f + expf(-acc));  // SiLU
  Y[(size_t)b * N + n] = acc;
}

__global__ void silu_kernel(const float* __restrict__ in, float* __restrict__ out, int n) {
  const int i = blockIdx.x * blockDim.x + threadIdx.x;
  if (i < n) {
    float v = in[i];
    out[i] = v / (1.f + expf(-v));
  }
}

// ---------------------------------------------------------------------------
// Host-side dispatch
// ---------------------------------------------------------------------------
static GemmArgs mk(const float* A, const float* Bm, const float* bias, float* D, int M,
                   int N, int K, int lda, int ldb, int ldd) {
  GemmArgs g{};
  g.A = A; g.Bm = Bm; g.bias = bias; g.D = D;
  g.M = M; g.N = N; g.K = K;
  g.lda = lda; g.ldb = ldb; g.ldd = ldd;
  g.zdiv = 1; g.alpha = 1.f; g.rowsPerB = TOK_; g.gateStride = 0;
  return g;
}

template <bool WNK, int EPI>
static void rungemm(const GemmArgs& g, int Z, int threads, hipStream_t s) {
  const int ntiles = g.N >> 4;
  const int perWG = (threads >> 5) * TPW_;
  const int gy = (ntiles + perWG - 1) / perWG;
  dim3 grid(g.M >> 4, gy, Z);
  gemm_kernel<WNK, EPI><<<grid, threads, 0, s>>>(g);
}

extern "C" void kernel_launch(void* const* d_in, const int* in_sizes, int n_in,
                              void* d_out, int out_size, void* d_ws, size_t ws_size,
                              hipStream_t stream) {
  const float* x = (const float*)d_in[0];
  const float* tt = (const float*)d_in[1];
  const float* coords = (const float*)d_in[2];
  const float* pln1w = (const float*)d_in[3];
  const float* pln1b = (const float*)d_in[4];
  const float* projw = (const float*)d_in[5];
  const float* projb = (const float*)d_in[6];
  const float* pln2w = (const float*)d_in[7];
  const float* pln2b = (const float*)d_in[8];
  const float* tew1 = (const float*)d_in[9];
  const float* teb1 = (const float*)d_in[10];
  const float* tew2 = (const float*)d_in[11];
  const float* teb2 = (const float*)d_in[12];
  const float* bqkvw = (const float*)d_in[13];
  const float* bqkvb = (const float*)d_in[14];
  const float* bprojw = (const float*)d_in[15];
  const float* bprojb = (const float*)d_in[16];
  const float* bp1w = (const float*)d_in[17];
  const float* bp1b = (const float*)d_in[18];
  const float* bp1gw = (const float*)d_in[19];
  const float* bp1gb = (const float*)d_in[20];
  const float* bp2w = (const float*)d_in[21];
  const float* bp2b = (const float*)d_in[22];
  const float* bp2gw = (const float*)d_in[23];
  const float* bp2gb = (const float*)d_in[24];
  const float* bf1w = (const float*)d_in[25];
  const float* bf1b = (const float*)d_in[26];
  const float* bf2w = (const float*)d_in[27];
  const float* bf2b = (const float*)d_in[28];
  const float* badaw = (const float*)d_in[29];
  const float* badab = (const float*)d_in[30];
  const float* finaw = (const float*)d_in[31];
  const float* finab = (const float*)d_in[32];
  const float* finw = (const float*)d_in[33];
  const float* finb = (const float*)d_in[34];

  float* ws = (float*)d_ws;
  float* pe = ws;                    // 1,048,576
  float* xs = pe + 1048576;          // 1,048,576
  float* hh = xs + 1048576;          // 1,048,576
  float* t1 = hh + 1048576;          // 1,048,576
  float* t2 = t1 + 1048576;          // 1,048,576
  float* mask = t2 + 1048576;        //   524,288
  float* qkv = mask + 524288;        // 3,145,728
  float* S = qkv + 3145728;          // 4,194,304
  float* attno = S + 4194304;        // 1,048,576
  float* mlp = attno + 1048576;      // 4,194,304
  float* h0 = mlp + 4194304;         //   524,288
  float* cvec = h0 + 524288;         //     4,096
  float* scs = cvec + 4096;          //     4,096
  float* emb = scs + 4096;           //     2,048
  float* teh = emb + 2048;           //     4,096
  float* ada = teh + 4096;           //    24,576
  float* fina = ada + 24576;         //     8,192

  const int M = B_ * TOK_;  // 2048

  // positional encoding + input projection
  pe_kernel<<<B_ * TOK_, 256, 0, stream>>>(coords, pe);
  ln_x_kernel<<<B_ * TOK_, 128, 0, stream>>>(x, pln1w, pln1b, h0);
  {
    GemmArgs g = mk(h0, projw, projb, hh, M, LAT_, HID_, HID_, HID_, LAT_);
    rungemm<true, E_NONE>(g, 1, 256, stream);
  }
  ln_proj2_addpe_kernel<<<M, 256, 0, stream>>>(hh, pln2w, pln2b, pe, xs);

  // timestep embedding
  temb_kernel<<<B_, 128, 0, stream>>>(tt, emb);
  small_gemm_kernel<<<dim3(LAT_ / 256, B_), 256, 0, stream>>>(emb, tew1, teb1, teh,
                                                              LAT_, FREQ_, 1);
  small_gemm_kernel<<<dim3(LAT_ / 256, B_), 256, 0, stream>>>(teh, tew2, teb2, cvec,
                                                              LAT_, LAT_, 0);
  silu_kernel<<<(B_ * LAT_ + 255) / 256, 256, 0, stream>>>(cvec, scs, B_ * LAT_);

  for (int d = 0; d < DEPTH_; ++d) {
    const float* qw = bqkvw + (size_t)d * 3 * LAT_ * LAT_;
    const float* qb = bqkvb + (size_t)d * 3 * LAT_;
    const float* pw = bprojw + (size_t)d * LAT_ * LAT_;
    const float* pb = bprojb + (size_t)d * LAT_;
    const float* p1w = bp1w + (size_t)d * LAT_ * LAT_;
    const float* p1b = bp1b + (size_t)d * LAT_;
    const float* p1gw = bp1gw + (size_t)d * LAT_;
    const float* p1gb = bp1gb + (size_t)d * LAT_;
    const float* p2w = bp2w + (size_t)d * LAT_ * LAT_;
    const float* p2b = bp2b + (size_t)d * LAT_;
    const float* p2gw = bp2gw + (size_t)d * LAT_;
    const float* p2gb = bp2gb + (size_t)d * LAT_;
    const float* f1w = bf1w + (size_t)d * MLPH_ * LAT_;
    const float* f1b = bf1b + (size_t)d * MLPH_;
    const float* f2w = bf2w + (size_t)d * LAT_ * MLPH_;
    const float* f2b = bf2b + (size_t)d * LAT_;
    const float* aw = badaw + (size_t)d * 6 * LAT_ * LAT_;
    const float* ab = badab + (size_t)d * 6 * LAT_;

    // adaLN modulation vector: ada(B,3072) = silu(c) @ aw^T + ab
    small_gemm_kernel<<<dim3(6 * LAT_ / 256, B_), 256, 0, stream>>>(scs, aw, ab, ada,
                                                                    6 * LAT_, LAT_, 0);
    // hh = LN(xs)*(1+sc_a)+sh_a
    modln_kernel<<<M, 256, 0, stream>>>(xs, ada, 6 * LAT_, 0, LAT_, hh);

    // pe projections + GroupNorm (in-place)
    {
      GemmArgs g = mk(pe, p1w, p1b, t1, M, LAT_, LAT_, LAT_, LAT_, LAT_);
      rungemm<true, E_NONE>(g, 1, 256, stream);
    }
    gn_kernel<<<B_ * GROUPS_, 256, 0, stream>>>(t1, p1gw, p1gb, t1);
    {
      GemmArgs g = mk(pe, p2w, p2b, t2, M, LAT_, LAT_, LAT_, LAT_, LAT_);
      rungemm<true, E_NONE>(g, 1, 256, stream);
    }
    gn_kernel<<<B_ * GROUPS_, 256, 0, stream>>>(t2, p2gw, p2gb, t2);

    // mask[b] = sigmoid(pe1n[b] @ pe2n[b]^T)
    {
      GemmArgs g = mk(t1, t2, nullptr, mask, TOK_, TOK_, LAT_, LAT_, LAT_, TOK_);
      g.aO = (long)TOK_ * LAT_;
      g.bO = (long)TOK_ * LAT_;
      g.dO = (long)TOK_ * TOK_;
      rungemm<true, E_SIG>(g, B_, 256, stream);
    }

    // qkv = hh @ qw^T + qb
    {
      GemmArgs g = mk(hh, qw, qb, qkv, M, 3 * LAT_, LAT_, LAT_, LAT_, 3 * LAT_);
      rungemm<true, E_NONE>(g, 1, 256, stream);
    }

    // S[b,h] = (q @ k^T) * scale * mask[b]
    {
      GemmArgs g = mk(qkv, qkv + LAT_, nullptr, S, TOK_, TOK_, HD_, 3 * LAT_,
                      3 * LAT_, TOK_);
      g.zdiv = HEADS_;
      g.aO = (long)TOK_ * 3 * LAT_; g.aI = HD_;
      g.bO = (long)TOK_ * 3 * LAT_; g.bI = HD_;
      g.dO = (long)HEADS_ * TOK_ * TOK_; g.dI = (long)TOK_ * TOK_;
      g.mul = mask; g.mulO = (long)TOK_ * TOK_;
      g.alpha = 0.125f;  // HD^-0.5
      rungemm<true, E_MUL>(g, B_ * HEADS_, 256, stream);
    }
    softmax_kernel<<<B_ * HEADS_ * TOK_, 256, 0, stream>>>(S);

    // attno[b,:,h*64:...] = P @ V
    {
      GemmArgs g = mk(S, qkv + 2 * LAT_, nullptr, attno, TOK_, HD_, TOK_, TOK_,
                      3 * LAT_, LAT_);
      g.zdiv = HEADS_;
      g.aO = (long)HEADS_ * TOK_ * TOK_; g.aI = (long)TOK_ * TOK_;
      g.bO = (long)TOK_ * 3 * LAT_; g.bI = HD_;
      g.dO = (long)TOK_ * LAT_; g.dI = HD_;
      rungemm<false, E_NONE>(g, B_ * HEADS_, 128, stream);
    }

    // xs += g_a * (attno @ pw^T + pb)
    {
      GemmArgs g = mk(attno, pw, pb, xs, M, LAT_, LAT_, LAT_, LAT_, LAT_);
      g.gate = ada + 2 * LAT_; g.gateStride = 6 * LAT_; g.res = xs;
      rungemm<true, E_GATED>(g, 1, 256, stream);
    }

    // MLP with gated residual
    modln_kernel<<<M, 256, 0, stream>>>(xs, ada, 6 * LAT_, 3 * LAT_, 4 * LAT_, hh);
    {
      GemmArgs g = mk(hh, f1w, f1b, mlp, M, MLPH_, LAT_, LAT_, LAT_, MLPH_);
      rungemm<true, E_GELU>(g, 1, 256, stream);
    }
    {
      GemmArgs g = mk(mlp, f2w, f2b, xs, M, LAT_, MLPH_, MLPH_, MLPH_, LAT_);
      g.gate = ada + 5 * LAT_; g.gateStride = 6 * LAT_; g.res = xs;
      rungemm<true, E_GATED>(g, 1, 256, stream);
    }
  }

  // FinalLayer
  small_gemm_kernel<<<dim3(2 * LAT_ / 256, B_), 256, 0, stream>>>(scs, finaw, finab,
                                                                  fina, 2 * LAT_, LAT_, 0);
  modln_kernel<<<M, 256, 0, stream>>>(xs, fina, 2 * LAT_, 0, LAT_, hh);
  {
    GemmArgs g = mk(hh, finw, finb, (float*)d_out, M, OUTC_, LAT_, LAT_, LAT_, 0);
    g.rowsPerB = TOK_;
    rungemm<true, E_TOUT>(g, 1, 256, stream);
  }
}